// CFLiner_46729244180786
// MI455X (gfx1250) — compile-verified
//
#include <hip/hip_runtime.h>

// Clifford (2^8) linear layer on MI455X (gfx1250, wave32):
//   Out[16384 x 4096] = A[16384 x 256] x Beff[256 x 4096] + bias
//   Beff[i, k*256+c]  = sign(i, i^c) * W[k, i^c]
// Kernel 1 builds Beff (1M elems) in d_ws; Kernel 2 is a WMMA fp32 GEMM.
//
// LDS B tile is K-pair interleaved (+rotate-16 swizzle on odd k-pairs):
//   phys dword = (p*64 + ((n + (p&1)*16) & 63))*2 + (k&1),  p = k>>1
// so each WMMA B fragment is a single aligned ds_load_b64 (no repack movs)
// and half-wave bank conflicts are eliminated.

typedef __attribute__((ext_vector_type(2))) float v2f;
typedef __attribute__((ext_vector_type(8))) float v8f;

#define SDIM   256     // blade dimension
#define KCH    16      // out channels
#define NTOT   4096    // KCH * SDIM (GEMM N)
#define NBATCH 16384   // GEMM M
#define MBLK   128     // block M tile (8 waves x 16)
#define NBLK   64      // block N tile (4 WMMA tiles per wave)

// ---------------------------------------------------------------------------
// Kernel 1: Beff[i, n] = sign(i, i^c) * W[k, i^c],  n = k*256 + c
// ---------------------------------------------------------------------------
__global__ __launch_bounds__(256) void cf_build_beff(const float* __restrict__ W,
                                                     float* __restrict__ Beff) {
    const int n = blockIdx.x * blockDim.x + threadIdx.x;  // 0..4095
    const int i = blockIdx.y;                             // 0..255
    const int c = n & (SDIM - 1);
    const int k = n >> 8;
    const int j = i ^ c;

    int d   = __popc(i);
    int neg = 0;
    const int p = (i > j) ? i : j;
    for (int e = 1; e <= p; e <<= 1) {
        if (e & i) --d;
        if ((d & 1) && (e & j)) neg ^= 1;
    }
    const float sgn = neg ? -1.0f : 1.0f;
    Beff[i * NTOT + n] = sgn * W[k * SDIM + j];
}

// ---------------------------------------------------------------------------
// Kernel 2: fp32 GEMM via V_WMMA_F32_16X16X4_F32.
//  - 256 threads = 8 wave32 waves; block tile 128(M) x 64(N).
//  - Beff block tile (256 x 64 fp32 = 64 KB) staged in LDS, pair-interleaved.
//  - Each wave: 16(M) x 64(N) strip = 4 accumulator tiles; A-frag reused 4x.
//  - Output stores are non-temporal: 268 MB write-once stream must not evict
//    L2-resident A (16.8 MB, reused 64x) / Beff (4 MB, reused 128x).
// ---------------------------------------------------------------------------
__global__ __launch_bounds__(256) void cf_gemm_wmma(const float* __restrict__ A,
                                                    const float* __restrict__ Beff,
                                                    const float* __restrict__ bias,
                                                    float* __restrict__ Out) {
    __shared__ float Bs[SDIM * NBLK];  // 64 KB, pair-interleaved + swizzled

    const int tid   = threadIdx.x;
    const int wave  = tid >> 5;
    const int lane  = tid & 31;
    const int n16   = lane & 15;   // N (or M) position within a 16-wide tile
    const int khalf = lane >> 4;   // 0 -> K pair {kb,kb+1}; 1 -> {kb+2,kb+3}

    const int colBase = blockIdx.x * NBLK;
    const int rowBase = blockIdx.y * MBLK;
    const int waveRow = rowBase + wave * 16;

    // --- Cooperative fill: pack {B[2p][n..n+1], B[2p+1][n..n+1]} into one
    //     interleaved float4 per (k-pair, col-pair): ds_store_b128.
    {
        const int nitems = (SDIM / 2) * (NBLK / 2);  // 128 * 32 = 4096
        for (int q = tid; q < nitems; q += 256) {
            const int p  = q >> 5;        // k-pair 0..127
            const int nn = (q & 31) << 1; // even col 0..62
            const v2f r0 = *(const v2f*)(Beff + (size_t)(2 * p    ) * NTOT + colBase + nn);
            const v2f r1 = *(const v2f*)(Beff + (size_t)(2 * p + 1) * NTOT + colBase + nn);
            const int nsw = (nn + ((p & 1) << 4)) & (NBLK - 1);  // swizzle
            float4 pk;
            pk.x = r0.x; pk.y = r1.x; pk.z = r0.y; pk.w = r1.y;
            *(float4*)(Bs + (((p << 6) + nsw) << 1)) = pk;
        }
    }
    __syncthreads();

    v8f acc0 = {}, acc1 = {}, acc2 = {}, acc3 = {};

    // Per-lane A pointer: row (waveRow + n16), starting K = 2*khalf.
    const float* Aptr = A + (size_t)(waveRow + n16) * SDIM + 2 * khalf;

    // Loop-invariant per-lane B dword indices (k-pair p = kb/2 + khalf,
    // p&1 == khalf since kb % 4 == 0). Swizzled col = ((t+khalf)&3)*16 + n16.
    const int idx0 = (khalf << 7) + (((((0 + khalf) & 3) << 4) + n16) << 1);
    const int idx1 = (khalf << 7) + (((((1 + khalf) & 3) << 4) + n16) << 1);
    const int idx2 = (khalf << 7) + (((((2 + khalf) & 3) << 4) + n16) << 1);
    const int idx3 = (khalf << 7) + (((((3 + khalf) & 3) << 4) + n16) << 1);

#pragma unroll 4
    for (int kb = 0; kb < SDIM; kb += 4) {
        const v2f a = *(const v2f*)(Aptr + kb);      // global_load_b64

        const float* bk = Bs + kb * (NBLK);          // (kb/2)*128 dwords
        const v2f b0 = *(const v2f*)(bk + idx0);     // ds_load_b64, aligned pair
        const v2f b1 = *(const v2f*)(bk + idx1);
        const v2f b2 = *(const v2f*)(bk + idx2);
        const v2f b3 = *(const v2f*)(bk + idx3);

        acc0 = __builtin_amdgcn_wmma_f32_16x16x4_f32(false, a, false, b0,
                                                     (short)0, acc0, false, false);
        acc1 = __builtin_amdgcn_wmma_f32_16x16x4_f32(false, a, false, b1,
                                                     (short)0, acc1, false, false);
        acc2 = __builtin_amdgcn_wmma_f32_16x16x4_f32(false, a, false, b2,
                                                     (short)0, acc2, false, false);
        acc3 = __builtin_amdgcn_wmma_f32_16x16x4_f32(false, a, false, b3,
                                                     (short)0, acc3, false, false);
    }

    // Bias: each lane's column within a tile is fixed; c = col & 255.
    const float bv0 = bias[(colBase +  0 + n16) & (SDIM - 1)];
    const float bv1 = bias[(colBase + 16 + n16) & (SDIM - 1)];
    const float bv2 = bias[(colBase + 32 + n16) & (SDIM - 1)];
    const float bv3 = bias[(colBase + 48 + n16) & (SDIM - 1)];

    // C/D layout: VGPR r -> M = r + khalf*8, N = n16 (per 16-wide tile).
    const int rowOff = khalf * 8;
    float* outBase = Out + (size_t)(waveRow + rowOff) * NTOT + colBase + n16;
#pragma unroll
    for (int r = 0; r < 8; ++r) {
        float* o = outBase + (size_t)r * NTOT;
        __builtin_nontemporal_store(acc0[r] + bv0, o +  0);
        __builtin_nontemporal_store(acc1[r] + bv1, o + 16);
        __builtin_nontemporal_store(acc2[r] + bv2, o + 32);
        __builtin_nontemporal_store(acc3[r] + bv3, o + 48);
    }
}

// ---------------------------------------------------------------------------
extern "C" void kernel_launch(void* const* d_in, const int* in_sizes, int n_in,
                              void* d_out, int out_size, void* d_ws, size_t ws_size,
                              hipStream_t stream) {
    (void)in_sizes; (void)n_in; (void)out_size; (void)ws_size;

    const float* inp  = (const float*)d_in[0];  // [16384, 256]
    const float* w    = (const float*)d_in[1];  // [16, 256]
    const float* bias = (const float*)d_in[2];  // [256]
    float* out  = (float*)d_out;                // [16384, 16, 256]
    float* Beff = (float*)d_ws;                 // 256*4096 fp32 = 4 MB scratch

    cf_build_beff<<<dim3(NTOT / 256, SDIM), 256, 0, stream>>>(w, Beff);
    cf_gemm_wmma<<<dim3(NTOT / NBLK, NBATCH / MBLK), 256, 0, stream>>>(inp, Beff, bias, out);
}